// FragAttention_8933531976300
// MI455X (gfx1250) — compile-verified
//
#include <hip/hip_runtime.h>
#include <hip/hip_bf16.h>

typedef __attribute__((ext_vector_type(2))) float v2f;
typedef __attribute__((ext_vector_type(8))) float v8f;

#define Sdim 128
#define Bdim 512
#define Ddim 512
#define Gdim 127
#define TN   128          // d-columns per workgroup
#define AP   132          // padded stride for Aw rows (g -> s), mult of 4
#define SP   132          // padded stride for XsT rows (d -> s), mult of 4

__global__ __launch_bounds__(256)
void frag_prefix_wmma(const float* __restrict__ x,
                      const unsigned char* __restrict__ src_mask,
                      float* __restrict__ out)
{
    // Aw[g][s] = (s <= g) ? valid[s] : 0   (masked triangular weights, per b)
    __shared__ __align__(16) float Aw[Sdim * AP];
    // XsT[d][s] = x[s, b, dt*128 + d]      (transposed so K is contiguous)
    __shared__ __align__(16) float XsT[TN * SP];
    __shared__ float validf[Sdim];
    __shared__ float totals[TN];

    const int tid = threadIdx.x;
    const int dt  = blockIdx.x;        // 0..3   (d tile)
    const int b   = blockIdx.y;        // 0..511

    if (tid < Sdim)
        validf[tid] = src_mask[b * Sdim + tid] ? 0.0f : 1.0f;
    __syncthreads();

    // Build masked triangular A in LDS: 16384 elems, 64 per thread.
    for (int i = tid; i < Sdim * Sdim; i += 256) {
        const int g = i >> 7;
        const int s = i & (Sdim - 1);
        Aw[g * AP + s] = (s <= g) ? validf[s] : 0.0f;
    }

    // Stage X tile transposed: read float4 along d (coalesced), scatter into XsT.
    {
        const int nvec = (Sdim * TN) / 4;              // 4096 float4
        for (int i = tid; i < nvec; i += 256) {
            const int s  = i >> 5;                     // i / 32
            const int d0 = (i & 31) << 2;              // column base (float4)
            const float4 v = *(const float4*)(x + ((size_t)s * Bdim + b) * Ddim
                                                + (size_t)dt * TN + d0);
            XsT[(d0 + 0) * SP + s] = v.x;
            XsT[(d0 + 1) * SP + s] = v.y;
            XsT[(d0 + 2) * SP + s] = v.z;
            XsT[(d0 + 3) * SP + s] = v.w;
        }
    }
    __syncthreads();

    const int wave = tid >> 5;
    const int lane = tid & 31;
    const int half = lane >> 4;        // 0: lanes 0-15 (K=0,1), 1: lanes 16-31 (K=2,3)
    const int lm   = lane & 15;
    const int m0   = wave * 16;        // this wave's M strip
    const int Mrow = m0 + lm;          // A-matrix row held by this lane

    const float* aRow = &Aw[Mrow * AP];

    v8f acc[8] = {};                   // 8 tiles of 16x16 f32 accumulators

    // K loop: 32 steps of K=4 over s = 0..127.  Pure ds_load_b64 + wmma.
    for (int k0 = 0; k0 < Sdim; k0 += 4) {
        const int s0 = k0 + half * 2;                 // even -> 8B aligned
        const v2f a = *(const v2f*)(aRow + s0);       // A: K=s0, s0+1
        #pragma unroll
        for (int t = 0; t < 8; ++t) {
            const int col = t * 16 + lm;              // B-matrix N index
            const v2f bb = *(const v2f*)(&XsT[col * SP + s0]);  // B: K=s0, s0+1
            acc[t] = __builtin_amdgcn_wmma_f32_16x16x4_f32(
                false, a, false, bb, (short)0, acc[t], false, false);
        }
    }

    // Row g = 127 (all s kept) == column totals; lives in wave 7, lanes 16-31, elem 7.
    if (wave == 7 && half == 1) {
        #pragma unroll
        for (int t = 0; t < 8; ++t)
            totals[t * 16 + lm] = acc[t][7];
    }
    __syncthreads();

    // Emit left = prefix sum, right = total - left.  out shape (B, 127, 1024).
    #pragma unroll
    for (int t = 0; t < 8; ++t) {
        const int col   = t * 16 + lm;
        const int dglob = dt * TN + col;
        const float tot = totals[col];
        #pragma unroll
        for (int v = 0; v < 8; ++v) {
            const int g = m0 + half * 8 + v;           // C/D layout: VGPR v, half-wave row
            if (g < Gdim) {
                const size_t o = ((size_t)b * Gdim + g) * (size_t)(2 * Ddim) + dglob;
                const float left = acc[t][v];
                out[o]        = left;
                out[o + Ddim] = tot - left;
            }
        }
    }
}

extern "C" void kernel_launch(void* const* d_in, const int* in_sizes, int n_in,
                              void* d_out, int out_size, void* d_ws, size_t ws_size,
                              hipStream_t stream) {
    (void)in_sizes; (void)n_in; (void)out_size; (void)d_ws; (void)ws_size;
    const float*         x    = (const float*)d_in[0];
    const unsigned char* mask = (const unsigned char*)d_in[1];  // jnp bool: 1 byte/elem
    float*               out  = (float*)d_out;

    dim3 grid(Ddim / TN, Bdim);   // (4, 512)
    frag_prefix_wmma<<<grid, 256, 0, stream>>>(x, mask, out);
}